// GTEProgramClassification_27986006900854
// MI455X (gfx1250) — compile-verified
//
#include <hip/hip_runtime.h>

// ---------------------------------------------------------------------------
// Fused Child-Sum Tree-LSTM step + classifier for gfx1250 (MI455X).
// Exploits ch_c == 0 (reference pads node_feat with zeros) and dead LN(c_out).
// fp32 WMMA (V_WMMA_F32_16X16X4_F32) keeps reference precision.
// ---------------------------------------------------------------------------

typedef __attribute__((ext_vector_type(2))) float v2f;
typedef __attribute__((ext_vector_type(8))) float v8f;

constexpr int kD       = 128;    // feature dim
constexpr int kL       = 8;      // mailbox slots
constexpr int kNDst    = 50000;  // destination nodes
constexpr int kClasses = 104;
constexpr int kRows    = 32;     // dest rows per block (2 WMMA M-tiles)
constexpr int kLda     = 132;    // padded LDS row stride (floats) -> no bank conflicts
constexpr int kGld     = 384;    // gates LDS row stride (i|o|u concat)

__device__ __forceinline__ float sigmoidf_(float x) { return 1.0f / (1.0f + __expf(-x)); }

__global__ __launch_bounds__(128)
void treelstm_fused_kernel(const int*   __restrict__ token_ids,
                           const int*   __restrict__ mailbox,
                           const float* __restrict__ emb,
                           const float* __restrict__ ix_w, const float* __restrict__ ih_w,
                           const float* __restrict__ ux_w, const float* __restrict__ uh_w,
                           const float* __restrict__ ox_w, const float* __restrict__ oh_w,
                           const float* __restrict__ ix_b, const float* __restrict__ ih_b,
                           const float* __restrict__ ux_b, const float* __restrict__ uh_b,
                           const float* __restrict__ ox_b, const float* __restrict__ oh_b,
                           const float* __restrict__ ln2_g, const float* __restrict__ ln2_b,
                           const float* __restrict__ fc_w,  const float* __restrict__ fc_b,
                           float* __restrict__ out)
{
    __shared__ float sAx[kRows * kLda];   // x tile      (32 x 128, padded)
    __shared__ float sAh[kRows * kLda];   // h_sum tile
    __shared__ float sG [kRows * kGld];   // gate pre-activations (i|o|u)
    __shared__ float sH [kRows * kLda];   // h (then layer-normed h)
    __shared__ int   sSrc[kRows * kL];    // resolved source rows of emb

    const int t      = threadIdx.x;
    const int wave   = t >> 5;
    const int lane   = t & 31;
    const int m      = lane & 15;   // M index within 16-tile / N index for B frag
    const int g      = lane >> 4;   // lane group (K sub-offset)
    const int block0 = blockIdx.x * kRows;

    // ---- stage mailbox -> embedding row indices -------------------------------
    for (int i = t; i < kRows * kL; i += blockDim.x) {
        const int n = block0 + (i >> 3);
        int src = 0;
        if (n < kNDst) src = token_ids[mailbox[(long)n * kL + (i & 7)]];
        sSrc[i] = src;
    }
    __syncthreads();

    // ---- Phase A: gather x and child-sum h into LDS ---------------------------
    for (int r = wave; r < kRows; r += 4) {
        const int n = block0 + r;
        float4 hs = make_float4(0.f, 0.f, 0.f, 0.f);
        float4 xv = make_float4(0.f, 0.f, 0.f, 0.f);
        if (n < kNDst) {
            const int c = lane * 4;
#pragma unroll
            for (int j = 0; j < 7; ++j) {
                const float4 e = *(const float4*)&emb[(long)sSrc[r * kL + j] * kD + c];
                hs.x += e.x; hs.y += e.y; hs.z += e.z; hs.w += e.w;
            }
            xv = *(const float4*)&emb[(long)sSrc[r * kL + 7] * kD + c];
        }
        *(float4*)&sAx[r * kLda + lane * 4] = xv;
        *(float4*)&sAh[r * kLda + lane * 4] = hs;
    }
    __syncthreads();

    // ---- Phase B: gates GEMM: G[32x384] = X*Wx^T + H*Wh^T ---------------------
    // column tiles 0..7 -> i (ix/ih), 8..15 -> o (ox/oh), 16..23 -> u (ux/uh)
    for (int oc = wave * 6; oc < wave * 6 + 6; ++oc) {
        const float* wx = (oc < 8) ? ix_w : (oc < 16) ? ox_w : ux_w;
        const float* wh = (oc < 8) ? ih_w : (oc < 16) ? oh_w : uh_w;
        const int    ocl = (oc & 7) * 16;
        const float* wxr = wx + (long)(ocl + m) * kD;   // B[k][n] = W[ocl+n][k]
        const float* whr = wh + (long)(ocl + m) * kD;

        v8f c0 = {}; v8f c1 = {};
#pragma unroll 4
        for (int ks = 0; ks < 32; ++ks) {
            const int k = ks * 4 + g * 2;
            const v2f bx  = *(const v2f*)&wxr[k];
            const v2f bh  = *(const v2f*)&whr[k];
            const v2f ax0 = *(const v2f*)&sAx[m * kLda + k];
            const v2f ah0 = *(const v2f*)&sAh[m * kLda + k];
            const v2f ax1 = *(const v2f*)&sAx[(m + 16) * kLda + k];
            const v2f ah1 = *(const v2f*)&sAh[(m + 16) * kLda + k];
            c0 = __builtin_amdgcn_wmma_f32_16x16x4_f32(false, ax0, false, bx, (short)0, c0, false, false);
            c0 = __builtin_amdgcn_wmma_f32_16x16x4_f32(false, ah0, false, bh, (short)0, c0, false, false);
            c1 = __builtin_amdgcn_wmma_f32_16x16x4_f32(false, ax1, false, bx, (short)0, c1, false, false);
            c1 = __builtin_amdgcn_wmma_f32_16x16x4_f32(false, ah1, false, bh, (short)0, c1, false, false);
        }
        const int col = oc * 16 + m;
#pragma unroll
        for (int r = 0; r < 8; ++r) {
            sG[(r + g * 8) * kGld + col]      = c0[r];
            sG[(16 + r + g * 8) * kGld + col] = c1[r];
        }
    }
    __syncthreads();

    // ---- Phase C: pointwise gates -> h ----------------------------------------
    {
        const int d  = t;                      // blockDim.x == 128 == kD
        const float bi = ix_b[d] + ih_b[d];
        const float bo = ox_b[d] + oh_b[d];
        const float bu = ux_b[d] + uh_b[d];
        for (int n = 0; n < kRows; ++n) {
            const float iv = sigmoidf_(sG[n * kGld + d]       + bi);
            const float ov = sigmoidf_(sG[n * kGld + 128 + d] + bo);
            const float uv = tanhf    (sG[n * kGld + 256 + d] + bu);
            const float c  = iv * uv;
            sH[n * kLda + d] = ov * tanhf(c);
        }
    }
    __syncthreads();

    // ---- Phase C2: row-wise LayerNorm (wave32 shfl reduction) -----------------
    for (int r = wave; r < kRows; r += 4) {
        float4 hv = *(const float4*)&sH[r * kLda + lane * 4];
        float s  = hv.x + hv.y + hv.z + hv.w;
        float ss = hv.x * hv.x + hv.y * hv.y + hv.z * hv.z + hv.w * hv.w;
#pragma unroll
        for (int off = 16; off > 0; off >>= 1) {
            s  += __shfl_xor(s,  off, 32);
            ss += __shfl_xor(ss, off, 32);
        }
        const float mu  = s * (1.0f / 128.0f);
        const float var = ss * (1.0f / 128.0f) - mu * mu;
        const float rs  = rsqrtf(var + 1e-5f);
        const float4 gg = *(const float4*)&ln2_g[lane * 4];
        const float4 bb = *(const float4*)&ln2_b[lane * 4];
        hv.x = (hv.x - mu) * rs * gg.x + bb.x;
        hv.y = (hv.y - mu) * rs * gg.y + bb.y;
        hv.z = (hv.z - mu) * rs * gg.z + bb.z;
        hv.w = (hv.w - mu) * rs * gg.w + bb.w;
        *(float4*)&sH[r * kLda + lane * 4] = hv;
    }
    __syncthreads();

    // ---- Phase D: classifier GEMM: logits[32x104] = H * fc_w^T + fc_b ---------
    for (int oc = wave; oc < 7; oc += 4) {
        const int row  = oc * 16 + m;                 // class index (N of B)
        const int rrow = (row < kClasses) ? row : 0;  // clamp OOB lanes (unstored)
        const float* fr = fc_w + (long)rrow * kD;

        v8f c0 = {}; v8f c1 = {};
#pragma unroll 4
        for (int ks = 0; ks < 32; ++ks) {
            const int k = ks * 4 + g * 2;
            const v2f b  = *(const v2f*)&fr[k];
            const v2f a0 = *(const v2f*)&sH[m * kLda + k];
            const v2f a1 = *(const v2f*)&sH[(m + 16) * kLda + k];
            c0 = __builtin_amdgcn_wmma_f32_16x16x4_f32(false, a0, false, b, (short)0, c0, false, false);
            c1 = __builtin_amdgcn_wmma_f32_16x16x4_f32(false, a1, false, b, (short)0, c1, false, false);
        }
        if (row < kClasses) {
            const float bias = fc_b[row];
#pragma unroll
            for (int r = 0; r < 8; ++r) {
                const int n0 = block0 + r + g * 8;
                if (n0 < kNDst) out[(long)n0 * kClasses + row] = c0[r] + bias;
                const int n1 = block0 + 16 + r + g * 8;
                if (n1 < kNDst) out[(long)n1 * kClasses + row] = c1[r] + bias;
            }
        }
    }
}

extern "C" void kernel_launch(void* const* d_in, const int* in_sizes, int n_in,
                              void* d_out, int out_size, void* d_ws, size_t ws_size,
                              hipStream_t stream)
{
    (void)in_sizes; (void)n_in; (void)out_size; (void)d_ws; (void)ws_size;
    const int*   token_ids = (const int*)  d_in[0];
    const int*   mailbox   = (const int*)  d_in[1];
    const float* emb       = (const float*)d_in[2];
    const float* ix_w      = (const float*)d_in[3];
    const float* ih_w      = (const float*)d_in[4];
    // d_in[5] fx_w, d_in[6] fh_w: dead (multiplied by provably-zero ch_c)
    const float* ux_w      = (const float*)d_in[7];
    const float* uh_w      = (const float*)d_in[8];
    const float* ox_w      = (const float*)d_in[9];
    const float* oh_w      = (const float*)d_in[10];
    const float* ix_b      = (const float*)d_in[11];
    const float* ih_b      = (const float*)d_in[12];
    // d_in[13] fx_b, d_in[14] fh_b: dead
    const float* ux_b      = (const float*)d_in[15];
    const float* uh_b      = (const float*)d_in[16];
    const float* ox_b      = (const float*)d_in[17];
    const float* oh_b      = (const float*)d_in[18];
    // d_in[19] ln1_g, d_in[20] ln1_b: normalize c_out which is never used
    const float* ln2_g     = (const float*)d_in[21];
    const float* ln2_b     = (const float*)d_in[22];
    const float* fc_w      = (const float*)d_in[23];
    const float* fc_b      = (const float*)d_in[24];
    float* out = (float*)d_out;

    dim3 grid((kNDst + kRows - 1) / kRows);
    dim3 block(128);
    hipLaunchKernelGGL(treelstm_fused_kernel, grid, block, 0, stream,
                       token_ids, mailbox, emb,
                       ix_w, ih_w, ux_w, uh_w, ox_w, oh_w,
                       ix_b, ih_b, ux_b, uh_b, ox_b, oh_b,
                       ln2_g, ln2_b, fc_w, fc_b, out);
}